// Quantize_2_12756052869865
// MI455X (gfx1250) — compile-verified
//
#include <hip/hip_runtime.h>
#include <hip/hip_bf16.h>

// Problem constants (from reference): B=32, T=512, DIM=256, N_EMBED=8192
#define NROWS   16384        // B*T
#define KDIM    256
#define NEMB    8192
#define QELEMS  (NROWS * KDIM)   // 4194304 floats of quantize output

typedef float v2f __attribute__((ext_vector_type(2)));
typedef float v8f __attribute__((ext_vector_type(8)));
typedef float f4  __attribute__((ext_vector_type(4)));

// DS-only workgroup barrier: orders LDS traffic without draining LOADcnt,
// so global prefetch loads stay in flight across the WMMA k-loop.
__device__ __forceinline__ void lds_barrier() {
    asm volatile("s_wait_dscnt 0\n\t"
                 "s_barrier_signal -1\n\t"
                 "s_barrier_wait -1" ::: "memory");
}

// ---------------------------------------------------------------------------
// Kernel 1: enorm[m] = sum_k E[k][m]^2  (coalesced over m), also zero diff slot
// ---------------------------------------------------------------------------
__global__ void vq_enorm_kernel(const float* __restrict__ E,
                                float* __restrict__ enorm,
                                float* __restrict__ diffslot) {
    int m = blockIdx.x * blockDim.x + threadIdx.x;
    float s = 0.0f;
    #pragma unroll 8
    for (int k = 0; k < KDIM; ++k) {
        float v = E[(size_t)k * NEMB + m];
        s = fmaf(v, v, s);
    }
    enorm[m] = s;
    if (m == 0) *diffslot = 0.0f;
}

// ---------------------------------------------------------------------------
// Kernel 2: ET[m][k] = E[k][m]  (32x32 LDS-tiled transpose, coalesced both ways)
// grid: (NEMB/32, KDIM/32), block: (32, 8)
// ---------------------------------------------------------------------------
__global__ void vq_transpose_kernel(const float* __restrict__ E,
                                    float* __restrict__ ET) {
    __shared__ float tile[32][33];
    int x0 = blockIdx.x * 32;   // n_embed dim
    int y0 = blockIdx.y * 32;   // k dim
    int tx = threadIdx.x, ty = threadIdx.y;
    #pragma unroll
    for (int j = 0; j < 32; j += 8)
        tile[ty + j][tx] = E[(size_t)(y0 + ty + j) * NEMB + x0 + tx];
    __syncthreads();
    #pragma unroll
    for (int j = 0; j < 32; j += 8)
        ET[(size_t)(x0 + ty + j) * KDIM + y0 + tx] = tile[tx][ty + j];
}

// ---------------------------------------------------------------------------
// Kernel 3: WMMA distance GEMM + per-row argmin.
// Block = 256 threads (8 waves) covering 128 rows (one 16-row tile per wave).
// Sweeps all 8192 columns in 16-wide tiles; E tiles are double-buffered in LDS
// with register prefetch so the global (L2-resident) fetch of tile i+1 runs
// concurrently with the 64-deep WMMA k-loop on tile i. Two independent
// accumulator chains break the WMMA D->C dependency so matrix ops can issue
// back-to-back.
// score = ||e||^2 - 2 * (x . e)   (row-constant ||x||^2 omitted for argmin)
// ---------------------------------------------------------------------------
#define XROWS    128
#define XSTRIDE  260          // 256 + 4 pad: avoids 64-bank conflicts on A reads
#define NTILES   (NEMB / 16)  // 512 column tiles

__global__ void __launch_bounds__(256, 1)
vq_argmin_kernel(const float* __restrict__ X,      // [NROWS, KDIM]
                 const float* __restrict__ E,      // [KDIM, NEMB]
                 const float* __restrict__ enorm,  // [NEMB]
                 int* __restrict__ outIdx) {       // [NROWS]
    __shared__ float Xs[XROWS * XSTRIDE];   // 130.0 KB
    __shared__ float Es[2][KDIM * 16];      // 2 x 16.0 KB (double buffer)

    const int tid  = threadIdx.x;
    const int wave = tid >> 5;       // 0..7 -> row tile
    const int lane = tid & 31;
    const int half = lane >> 4;      // selects K{0,1} vs K{2,3} / rows +8
    const int lc   = lane & 15;      // column within tile / row within tile

    const f4* Eg4 = (const f4*)E;    // E row k = k*(NEMB/4) float4s

    // ---- prefetch E tile 0 into registers (4 x b128 per thread) ----
    f4 pre[4];
    #pragma unroll
    for (int u = 0; u < 4; ++u) {
        int j = tid + u * 256;             // float4 index within 1024-f4 tile
        int k = j >> 2, c4 = j & 3;
        pre[u] = Eg4[(size_t)k * (NEMB / 4) + c4];
    }

    // ---- stage 128 rows of X into LDS (float4, padded stride) ----
    const f4* Xg = (const f4*)(X + (size_t)blockIdx.x * XROWS * KDIM);
    for (int i = tid; i < XROWS * (KDIM / 4); i += 256) {
        int row = i >> 6;                  // KDIM/4 = 64 float4 per row
        int q   = i & 63;
        f4 v = Xg[row * (KDIM / 4) + q];
        *(f4*)&Xs[row * XSTRIDE + q * 4] = v;
    }
    lds_barrier();

    // per-lane running argmin state: 8 rows per lane (C-matrix layout)
    float minv[8];
    int   mini[8];
    #pragma unroll
    for (int j = 0; j < 8; ++j) { minv[j] = __builtin_huge_valf(); mini[j] = 0; }

    // A fragment base: row = wave*16 + lc, K offset 0 or 2 depending on half
    const float* Abase = &Xs[(wave * 16 + lc) * XSTRIDE + (half ? 2 : 0)];

    for (int it = 0; it < NTILES; ++it) {
        float* Eb = Es[it & 1];

        // commit prefetched tile `it` to LDS
        #pragma unroll
        for (int u = 0; u < 4; ++u) {
            int j = tid + u * 256;
            *(f4*)&Eb[j * 4] = pre[u];
        }
        // issue global prefetch of tile `it+1`; stays in flight across k-loop
        if (it + 1 < NTILES) {
            int cb4n = (it + 1) * 4;       // next tile col base in float4 units
            #pragma unroll
            for (int u = 0; u < 4; ++u) {
                int j = tid + u * 256;
                int k = j >> 2, c4 = j & 3;
                pre[u] = Eg4[(size_t)k * (NEMB / 4) + cb4n + c4];
            }
        }
        lds_barrier();   // single barrier/iteration is safe with double buffer

        const float* Bbase = &Eb[lc + (half ? 32 : 0)];
        // two independent accumulator chains: breaks WMMA D->C RAW chain
        v8f acc0 = {0.f, 0.f, 0.f, 0.f, 0.f, 0.f, 0.f, 0.f};
        v8f acc1 = {0.f, 0.f, 0.f, 0.f, 0.f, 0.f, 0.f, 0.f};
        #pragma unroll
        for (int k0 = 0; k0 < KDIM; k0 += 8) {
            v2f a0 = *(const v2f*)(Abase + k0);
            v2f b0;
            b0.x = Bbase[k0 * 16];
            b0.y = Bbase[k0 * 16 + 16];
            acc0 = __builtin_amdgcn_wmma_f32_16x16x4_f32(
                false, a0, false, b0, (short)0, acc0, false, false);

            v2f a1 = *(const v2f*)(Abase + k0 + 4);
            v2f b1;
            b1.x = Bbase[(k0 + 4) * 16];
            b1.y = Bbase[(k0 + 4) * 16 + 16];
            acc1 = __builtin_amdgcn_wmma_f32_16x16x4_f32(
                false, a1, false, b1, (short)0, acc1, false, false);
        }

        const int   col = it * 16 + lc;
        const float en  = enorm[col];
        #pragma unroll
        for (int j = 0; j < 8; ++j) {
            float s = fmaf(-2.0f, acc0[j] + acc1[j], en);
            if (s < minv[j]) { minv[j] = s; mini[j] = col; }
        }
    }

    // ---- butterfly reduce over the 16 lanes sharing each row group ----
    #pragma unroll
    for (int j = 0; j < 8; ++j) {
        #pragma unroll
        for (int off = 1; off < 16; off <<= 1) {
            float ov = __shfl_xor(minv[j], off, 32);
            int   oi = __shfl_xor(mini[j], off, 32);
            if (ov < minv[j] || (ov == minv[j] && oi < mini[j])) {
                minv[j] = ov; mini[j] = oi;
            }
        }
    }
    if (lc == 0) {
        #pragma unroll
        for (int j = 0; j < 8; ++j) {
            int row = blockIdx.x * XROWS + wave * 16 + j + (half ? 8 : 0);
            outIdx[row] = mini[j];
        }
    }
}

// ---------------------------------------------------------------------------
// Kernel 4: gather quantize rows + reduce diff = mean((q-x)^2)
// block 256 threads = one element per thread per row; 16 rows per block
// ---------------------------------------------------------------------------
__global__ void vq_gather_kernel(const float* __restrict__ X,
                                 const float* __restrict__ E,
                                 const float* __restrict__ ET,
                                 const int* __restrict__ idx,
                                 float* __restrict__ quant,
                                 float* __restrict__ diffslot,
                                 int useET) {
    const int t = threadIdx.x;   // element index within a row (0..255)
    float sum = 0.0f;
    #pragma unroll 4
    for (int rr = 0; rr < 16; ++rr) {
        int row = blockIdx.x * 16 + rr;
        int id  = idx[row];
        float q = useET ? ET[(size_t)id * KDIM + t]
                        : E[(size_t)t * NEMB + id];
        float x = X[(size_t)row * KDIM + t];
        quant[(size_t)row * KDIM + t] = q;
        float d = q - x;
        sum = fmaf(d, d, sum);
    }
    // wave then cross-wave reduction
    #pragma unroll
    for (int off = 16; off > 0; off >>= 1)
        sum += __shfl_down(sum, off, 32);
    __shared__ float red[8];
    int wave = t >> 5, lane = t & 31;
    if (lane == 0) red[wave] = sum;
    __syncthreads();
    if (t == 0) {
        float s = 0.0f;
        #pragma unroll
        for (int w = 0; w < 8; ++w) s += red[w];
        atomicAdd(diffslot, s * (1.0f / (float)QELEMS));
    }
}

// ---------------------------------------------------------------------------
extern "C" void kernel_launch(void* const* d_in, const int* in_sizes, int n_in,
                              void* d_out, int out_size, void* d_ws, size_t ws_size,
                              hipStream_t stream) {
    const float* X = (const float*)d_in[0];   // [32,512,256] f32
    const float* E = (const float*)d_in[1];   // [256,8192] f32
    // d_in[2] = fix (==1): EMA branch skipped in reference

    float* quant    = (float*)d_out;                 // [NROWS*KDIM]
    float* diffslot = (float*)d_out + QELEMS;        // scalar
    int*   ind      = (int*)((float*)d_out + QELEMS + 1); // [NROWS] int32 bits

    float* enorm = (float*)d_ws;                     // [NEMB]
    float* ET    = enorm + NEMB;                     // [NEMB*KDIM] (optional)
    const size_t needET = (size_t)(NEMB + (size_t)NEMB * KDIM) * sizeof(float);
    const int useET = (ws_size >= needET) ? 1 : 0;

    // 1) ||e||^2 per code + zero diff accumulator
    vq_enorm_kernel<<<NEMB / 256, 256, 0, stream>>>(E, enorm, diffslot);

    // 2) transpose codebook for coalesced gather (if workspace allows)
    if (useET) {
        dim3 g(NEMB / 32, KDIM / 32), b(32, 8);
        vq_transpose_kernel<<<g, b, 0, stream>>>(E, ET);
    }

    // 3) WMMA distance GEMM + argmin (double-buffered E tiles)
    vq_argmin_kernel<<<NROWS / XROWS, 256, 0, stream>>>(X, E, enorm, ind);

    // 4) gather winning codes + diff reduction
    vq_gather_kernel<<<NROWS / 16, 256, 0, stream>>>(X, E, ET, ind, quant,
                                                     diffslot, useET);
}